// BatchingCostModule_1047972020590
// MI455X (gfx1250) — compile-verified
//
#include <hip/hip_runtime.h>
#include <hip/hip_bf16.h>

// costs[n,m] = 0.5*(|x_n|^2 + |y_m|^2) - x_n . y_m
// N = M = 4096, D = 128, fp32 everywhere (matches reference precision).
//
// MI455X reasoning: output writes are 64 MB -> ~2.9us floor @ 23.3 TB/s, but
// the 4.3 GFLOP fp32 GEMM is the real limiter, so the kernel is built around
// back-to-back V_WMMA_F32_16X16X4_F32 with full K-reuse in LDS/registers.

typedef __attribute__((ext_vector_type(2))) float v2f;
typedef __attribute__((ext_vector_type(8))) float v8f;
typedef __attribute__((ext_vector_type(4))) float f4;

#define TILE 128
#define BK 32
// Row stride in floats: 36 -> 144B rows (16B aligned for float4 stores) and
// conflict-free 16-lane column reads (n*36 mod 64 distinct for n=0..15).
#define LDS_STRIDE (BK + 4)

// ---------------------------------------------------------------------------
// Half-norm kernel: hx[n] = 0.5*|x_n|^2, hy[m] = 0.5*|y_m|^2.
// One wave per row, lane-strided loads, shfl_xor tree reduction (wave32).
// ---------------------------------------------------------------------------
__global__ void half_norms_kernel(const float* __restrict__ x,
                                  const float* __restrict__ y,
                                  float* __restrict__ hx,
                                  float* __restrict__ hy,
                                  int N, int M, int D) {
  int wave = blockIdx.x * (blockDim.x >> 5) + (threadIdx.x >> 5);
  int lane = threadIdx.x & 31;
  if (wave >= N + M) return;
  const float* src = (wave < N) ? (x + (size_t)wave * D)
                                : (y + (size_t)(wave - N) * D);
  float s = 0.0f;
  for (int k = lane; k < D; k += 32) {
    float v = src[k];
    s += v * v;
  }
#pragma unroll
  for (int off = 16; off > 0; off >>= 1) s += __shfl_xor(s, off, 32);
  if (lane == 0) {
    if (wave < N) hx[wave] = 0.5f * s;
    else          hy[wave - N] = 0.5f * s;
  }
}

// ---------------------------------------------------------------------------
// Tiled cost-matrix GEMM: 128x128 tile per 256-thread block (8 waves).
// Wave grid 2x4: each wave computes a 64x32 subtile = 4x2 accumulators of
// V_WMMA_F32_16X16X4_F32. K processed in BK=32 chunks, double-buffered LDS.
// ---------------------------------------------------------------------------
__global__ __launch_bounds__(256, 1)
void cost_gemm_wmma_f32(const float* __restrict__ X,
                        const float* __restrict__ Y,
                        const float* __restrict__ hx,
                        const float* __restrict__ hy,
                        float* __restrict__ out,
                        int N, int M, int D) {
  __shared__ float Xs[2][TILE][LDS_STRIDE];
  __shared__ float Ys[2][TILE][LDS_STRIDE];

  const int t    = threadIdx.x;
  const int wave = t >> 5;        // 0..7
  const int lane = t & 31;
  const int half = lane >> 4;     // 0: lanes 0-15, 1: lanes 16-31
  const int l    = lane & 15;

  const int block_m = blockIdx.y * TILE;
  const int block_n = blockIdx.x * TILE;

  const int wm = (wave >> 2) * 64;  // wave row base within tile: 0 / 64
  const int wn = (wave & 3) * 32;   // wave col base within tile: 0..96

  v8f acc[4][2];
#pragma unroll
  for (int i = 0; i < 4; ++i)
#pragma unroll
    for (int j = 0; j < 2; ++j)
      acc[i][j] = (v8f){0.f, 0.f, 0.f, 0.f, 0.f, 0.f, 0.f, 0.f};

  // Staging registers for the software pipeline (4 float4 per tensor/thread).
  f4 rx[4], ry[4];

  // Each chunk: 128 rows x 32 floats = 1024 float4 per tensor; 4 per thread.
  auto load_chunk = [&](int k0) {
#pragma unroll
    for (int i = 0; i < 4; ++i) {
      int idx = t + i * 256;
      int row = idx >> 3;
      int c4  = idx & 7;
      rx[i] = *(const f4*)(X + (size_t)(block_m + row) * D + k0 + c4 * 4);
      ry[i] = *(const f4*)(Y + (size_t)(block_n + row) * D + k0 + c4 * 4);
    }
  };
  auto store_chunk = [&](int buf) {
#pragma unroll
    for (int i = 0; i < 4; ++i) {
      int idx = t + i * 256;
      int row = idx >> 3;
      int c4  = idx & 7;
      *(f4*)&Xs[buf][row][c4 * 4] = rx[i];
      *(f4*)&Ys[buf][row][c4 * 4] = ry[i];
    }
  };

  load_chunk(0);
  store_chunk(0);
  __syncthreads();

  const int nchunk = D / BK;  // 4 for D=128
  for (int c = 0; c < nchunk; ++c) {
    const int cur = c & 1;
    if (c + 1 < nchunk) load_chunk((c + 1) * BK);  // overlap global fetch

#pragma unroll
    for (int kk = 0; kk < BK; kk += 4) {
      // A frag (16x4 f32, 2 VGPRs): lanes 0-15 hold K=kk,kk+1 of row M=l;
      // lanes 16-31 hold K=kk+2,kk+3 (ISA 32-bit A-matrix layout).
      v2f a[4], b[2];
#pragma unroll
      for (int i = 0; i < 4; ++i)
        a[i] = *(const v2f*)&Xs[cur][wm + i * 16 + l][kk + half * 2];
      // B frag (4x16): B[k][n] = y[n][k], mirrored lane-half layout.
#pragma unroll
      for (int j = 0; j < 2; ++j)
        b[j] = *(const v2f*)&Ys[cur][wn + j * 16 + l][kk + half * 2];

#pragma unroll
      for (int i = 0; i < 4; ++i)
#pragma unroll
        for (int j = 0; j < 2; ++j)
          acc[i][j] = __builtin_amdgcn_wmma_f32_16x16x4_f32(
              /*neg_a=*/false, a[i], /*neg_b=*/false, b[j],
              /*c_mod=*/(short)0, acc[i][j],
              /*reuse_a=*/false, /*reuse_b=*/false);
    }

    if (c + 1 < nchunk) store_chunk((c + 1) & 1);
    __syncthreads();  // cover both LDS-read completion and next-buffer publish
  }

  // Epilogue: C/D layout — VGPR r: lanes 0-15 -> M=r, lanes 16-31 -> M=r+8.
#pragma unroll
  for (int i = 0; i < 4; ++i) {
    float hxv[8];
#pragma unroll
    for (int r = 0; r < 8; ++r)
      hxv[r] = hx[block_m + wm + i * 16 + r + half * 8];
#pragma unroll
    for (int j = 0; j < 2; ++j) {
      const int nn = block_n + wn + j * 16 + l;
      const float hyv = hy[nn];
#pragma unroll
      for (int r = 0; r < 8; ++r) {
        const int mm = block_m + wm + i * 16 + r + half * 8;
        out[(size_t)mm * M + nn] = hxv[r] + hyv - acc[i][j][r];
      }
    }
  }
}

extern "C" void kernel_launch(void* const* d_in, const int* in_sizes, int n_in,
                              void* d_out, int out_size, void* d_ws, size_t ws_size,
                              hipStream_t stream) {
  const float* x = (const float*)d_in[0];
  const float* y = (const float*)d_in[1];
  float* out = (float*)d_out;

  const int D = 128;
  const int N = in_sizes[0] / D;  // 4096
  const int M = in_sizes[1] / D;  // 4096

  float* hx = (float*)d_ws;       // N floats
  float* hy = hx + N;             // M floats

  // Half-norms: one wave per row, 8 waves per block.
  const int rows = N + M;
  const int norm_blocks = (rows + 7) / 8;
  half_norms_kernel<<<norm_blocks, 256, 0, stream>>>(x, y, hx, hy, N, M, D);

  dim3 grid(M / TILE, N / TILE);  // 32 x 32 blocks
  cost_gemm_wmma_f32<<<grid, 256, 0, stream>>>(x, y, hx, hy, out, N, M, D);
}